// MambaNet_83872121356671
// MI455X (gfx1250) — compile-verified
//
#include <hip/hip_runtime.h>

// Problem constants (match reference)
#define B_   8
#define L_   1080
#define DM   256       // D_MODEL
#define DI   512       // D_INNER
#define DS_  16        // D_STATE
#define DR   16        // DT_RANK
#define MLEN (B_ * L_) // 8640 token rows = 135 * 64

typedef float v2f __attribute__((ext_vector_type(2)));
typedef float v8f __attribute__((ext_vector_type(8)));

// -------- fp32 WMMA 16x16x4 (D = A(16x4) * B(4x16) + C) -----------------
__device__ __forceinline__ v8f wmma4(v2f a, v2f b, v8f c) {
  return __builtin_amdgcn_wmma_f32_16x16x4_f32(false, a, false, b, (short)0, c,
                                               false, false);
}

// =========================================================================
// Register-blocked GEMM:  C[m,n] (+)= sum_k A[m,k] * W[n,k]   (W is N x K)
// One wave computes a BM x BN block of 16x16 tiles (BM*BN WMMA per K-step,
// reusing BM A-fragments and BN B-fragments). ACT: 0=none, 1=softplus.
// =========================================================================
template <int BM, int BN, int ACT, bool LOADC, bool HASB>
__global__ __launch_bounds__(256) void gemm_wmma(
    const float* __restrict__ A, int lda, const float* __restrict__ W, int ldw,
    const float* __restrict__ bias, float* __restrict__ C, int ldc, int M,
    int N, int K) {
  int gw = (blockIdx.x * blockDim.x + threadIdx.x) >> 5;  // global wave id
  int lane = threadIdx.x & 31;
  int tNb = N / (16 * BN);
  int nBlk = (M / (16 * BM)) * tNb;
  if (gw >= nBlk) return;  // wave-uniform exit keeps EXEC all-ones for WMMA
  int bm = gw / tNb;
  int bn = gw - bm * tNb;
  int rc = lane & 15;  // A row within tile == C/D column within tile
  int kh = lane >> 4;  // k-pair selector

  const float* Ap[BM];
  const float* Wp[BN];
#pragma unroll
  for (int i = 0; i < BM; ++i)
    Ap[i] = A + (size_t)(bm * BM * 16 + i * 16 + rc) * lda + kh * 2;
#pragma unroll
  for (int j = 0; j < BN; ++j)
    Wp[j] = W + (size_t)(bn * BN * 16 + j * 16 + rc) * ldw + kh * 2;

  v8f zero = {};
  v8f acc[BM][BN];
#pragma unroll
  for (int i = 0; i < BM; ++i)
#pragma unroll
    for (int j = 0; j < BN; ++j) acc[i][j] = zero;

  for (int k = 0; k < K; k += 4) {
    v2f a[BM], b[BN];
#pragma unroll
    for (int i = 0; i < BM; ++i) a[i] = *(const v2f*)(Ap[i] + k);
#pragma unroll
    for (int j = 0; j < BN; ++j) b[j] = *(const v2f*)(Wp[j] + k);
#pragma unroll
    for (int i = 0; i < BM; ++i)
#pragma unroll
      for (int j = 0; j < BN; ++j) acc[i][j] = wmma4(a[i], b[j], acc[i][j]);
  }

#pragma unroll
  for (int i = 0; i < BM; ++i) {
#pragma unroll
    for (int j = 0; j < BN; ++j) {
      float bb = HASB ? bias[bn * BN * 16 + j * 16 + rc] : 0.0f;
      float* Cp = C + (size_t)(bm * BM * 16 + i * 16 + kh * 8) * ldc +
                  bn * BN * 16 + j * 16 + rc;
#pragma unroll
      for (int r = 0; r < 8; ++r) {
        float v = acc[i][j][r] + bb;
        if (LOADC) v += Cp[(size_t)r * ldc];                    // residual
        if (ACT == 1) v = (v > 20.0f) ? v : log1pf(__expf(v));  // softplus
        Cp[(size_t)r * ldc] = v;
      }
    }
  }
}

// =========================================================================
// Embed: h[m,c] = x[m,0]*W[0,c] + x[m,1]*W[1,c] + b[c]
// =========================================================================
__global__ __launch_bounds__(256) void embed_kernel(
    const float* __restrict__ x, const float* __restrict__ W,
    const float* __restrict__ b, float* __restrict__ o) {
  int idx = blockIdx.x * blockDim.x + threadIdx.x;
  if (idx >= MLEN * DM) return;
  int m = idx >> 8;
  int c = idx & 255;
  o[idx] = x[(size_t)m * 2] * W[c] + x[(size_t)m * 2 + 1] * W[DM + c] + b[c];
}

// =========================================================================
// Positional conv: circular 3-tap over L, per-tap GEMM over channels,
// register-blocked BM x BN like gemm_wmma.
// out[b,l,o] = sum_k sum_c h0[b,(l+k-1)%L,c] * pcW[o,c,k] + pcb[o]
// =========================================================================
template <int BM, int BN>
__global__ __launch_bounds__(256) void posconv_kernel(
    const float* __restrict__ h0, const float* __restrict__ pcW,
    const float* __restrict__ pcb, float* __restrict__ hout) {
  int gw = (blockIdx.x * blockDim.x + threadIdx.x) >> 5;
  int lane = threadIdx.x & 31;
  const int tNb = DM / (16 * BN);
  if (gw >= (MLEN / (16 * BM)) * tNb) return;
  int bm = gw / tNb;
  int bn = gw - bm * tNb;
  int rc = lane & 15;
  int kh = lane >> 4;

  int bidx[BM], lidx[BM];
#pragma unroll
  for (int i = 0; i < BM; ++i) {
    int m = bm * BM * 16 + i * 16 + rc;
    bidx[i] = m / L_;
    lidx[i] = m - bidx[i] * L_;
  }

  v8f zero = {};
  v8f acc[BM][BN];
#pragma unroll
  for (int i = 0; i < BM; ++i)
#pragma unroll
    for (int j = 0; j < BN; ++j) acc[i][j] = zero;

#pragma unroll
  for (int k = 0; k < 3; ++k) {
    const float* Ar[BM];
    const float* Wr[BN];
#pragma unroll
    for (int i = 0; i < BM; ++i) {
      int l2 = lidx[i] + k - 1;
      if (l2 < 0) l2 += L_;
      if (l2 >= L_) l2 -= L_;
      Ar[i] = h0 + ((size_t)bidx[i] * L_ + l2) * DM + kh * 2;
    }
#pragma unroll
    for (int j = 0; j < BN; ++j) {
      int ncol = bn * BN * 16 + j * 16 + rc;
      Wr[j] = pcW + (size_t)ncol * DM * 3 + kh * 6 + k;  // stride 3 in c
    }
    for (int c = 0; c < DM; c += 4) {
      v2f a[BM], bv[BN];
#pragma unroll
      for (int i = 0; i < BM; ++i) a[i] = *(const v2f*)(Ar[i] + c);
#pragma unroll
      for (int j = 0; j < BN; ++j) {
        bv[j].x = Wr[j][c * 3];
        bv[j].y = Wr[j][c * 3 + 3];
      }
#pragma unroll
      for (int i = 0; i < BM; ++i)
#pragma unroll
        for (int j = 0; j < BN; ++j) acc[i][j] = wmma4(a[i], bv[j], acc[i][j]);
    }
  }

#pragma unroll
  for (int i = 0; i < BM; ++i) {
#pragma unroll
    for (int j = 0; j < BN; ++j) {
      float bia = pcb[bn * BN * 16 + j * 16 + rc];
      float* Cp = hout + (size_t)(bm * BM * 16 + i * 16 + kh * 8) * DM +
                  bn * BN * 16 + j * 16 + rc;
#pragma unroll
      for (int r = 0; r < 8; ++r) Cp[(size_t)r * DM] = acc[i][j][r] + bia;
    }
  }
}

// =========================================================================
// LayerNorm over DM=256, one wave32 per row (8 elems/lane + shfl_xor tree)
// =========================================================================
__global__ __launch_bounds__(256) void ln_kernel(const float* __restrict__ h,
                                                 const float* __restrict__ g,
                                                 const float* __restrict__ b,
                                                 float* __restrict__ o) {
  int row = (blockIdx.x * blockDim.x + threadIdx.x) >> 5;
  int lane = threadIdx.x & 31;
  if (row >= MLEN) return;
  const float* xr = h + (size_t)row * DM;
  float v[8];
  float s = 0.f;
#pragma unroll
  for (int i = 0; i < 8; ++i) {
    v[i] = xr[lane + i * 32];
    s += v[i];
  }
#pragma unroll
  for (int off = 16; off; off >>= 1) s += __shfl_xor(s, off, 32);
  float mean = s * (1.0f / DM);
  float var = 0.f;
#pragma unroll
  for (int i = 0; i < 8; ++i) {
    float d = v[i] - mean;
    var += d * d;
  }
#pragma unroll
  for (int off = 16; off; off >>= 1) var += __shfl_xor(var, off, 32);
  float rstd = rsqrtf(var * (1.0f / DM) + 1e-5f);
  float* orow = o + (size_t)row * DM;
#pragma unroll
  for (int i = 0; i < 8; ++i) {
    int c = lane + i * 32;
    orow[c] = (v[i] - mean) * rstd * g[c] + b[c];
  }
}

// =========================================================================
// Depthwise causal conv (D_CONV=2) + SiLU.  u-half of xz has ld = 1024.
// =========================================================================
__global__ __launch_bounds__(256) void conv_silu_kernel(
    const float* __restrict__ xz, const float* __restrict__ cW,
    const float* __restrict__ cb, float* __restrict__ u) {
  int idx = blockIdx.x * blockDim.x + threadIdx.x;
  if (idx >= MLEN * DI) return;
  int m = idx >> 9;
  int c = idx & 511;
  int bb = m / L_;
  int l = m - bb * L_;
  float cur = xz[(size_t)m * (2 * DI) + c];
  float prev = (l > 0) ? xz[(size_t)(m - 1) * (2 * DI) + c] : 0.0f;
  float v = prev * cW[c * 2 + 0] + cur * cW[c * 2 + 1] + cb[c];
  u[idx] = v / (1.0f + __expf(-v));  // silu
}

// =========================================================================
// Selective scan, fused with u*D skip and silu(z) gating.
// One thread per (b, d) channel; 16-state recurrence in registers.
// Gated output overwrites the dt buffer (read-then-write, same thread/step).
// Next-step operands are prefetched (global_prefetch_b8) so the per-step
// loads overlap the 16 dependent v_exp ops.
// =========================================================================
__global__ __launch_bounds__(256) void scan_kernel(
    const float* __restrict__ u, float* dt /* in: dt, out: gated y */,
    const float* __restrict__ dbc, const float* __restrict__ xz,
    const float* __restrict__ A_log, const float* __restrict__ Dp) {
  int idx = blockIdx.x * blockDim.x + threadIdx.x;  // 0..4095
  int bb = idx >> 9;
  int d = idx & 511;
  float Av[DS_], st[DS_];
#pragma unroll
  for (int n = 0; n < DS_; ++n) {
    Av[n] = -__expf(A_log[d * DS_ + n]);
    st[n] = 0.0f;
  }
  float Dd = Dp[d];
  for (int t = 0; t < L_; ++t) {
    size_t r = (size_t)bb * L_ + t;
    // Prefetch next timestep (speculative; dropped silently if out of range)
    __builtin_prefetch(dt + (r + 1) * DI + d, 0, 1);
    __builtin_prefetch(u + (r + 1) * DI + d, 0, 1);
    __builtin_prefetch(dbc + (r + 1) * 48, 0, 1);
    __builtin_prefetch(xz + (r + 1) * (2 * DI) + DI + d, 0, 1);

    float dtv = dt[r * DI + d];
    float uv = u[r * DI + d];
    float du = dtv * uv;
    const float* bc = dbc + r * 48;
    float y = 0.0f;
#pragma unroll
    for (int n = 0; n < DS_; ++n) {
      st[n] = __expf(dtv * Av[n]) * st[n] + du * bc[16 + n];
      y += st[n] * bc[32 + n];
    }
    y += uv * Dd;
    float z = xz[r * (2 * DI) + DI + d];
    float sz = z / (1.0f + __expf(-z));
    dt[r * DI + d] = y * sz;  // gated output, overwrites consumed dt
  }
}

// =========================================================================
// Mean pool over L, then tiny fc (8x3 outputs)
// =========================================================================
__global__ __launch_bounds__(256) void pool_kernel(const float* __restrict__ h,
                                                   float* __restrict__ pooled) {
  int idx = blockIdx.x * blockDim.x + threadIdx.x;  // 0..2047
  if (idx >= B_ * DM) return;
  int bb = idx >> 8;
  int c = idx & 255;
  float s = 0.f;
  for (int l = 0; l < L_; ++l) s += h[((size_t)bb * L_ + l) * DM + c];
  pooled[idx] = s * (1.0f / L_);
}

__global__ void fc_kernel(const float* __restrict__ pooled,
                          const float* __restrict__ fcW,
                          const float* __restrict__ fcb,
                          float* __restrict__ out) {
  int idx = threadIdx.x;
  if (idx >= B_ * 3) return;
  int bb = idx / 3;
  int o = idx - bb * 3;
  float s = 0.f;
  for (int d = 0; d < DM; ++d) s += pooled[bb * DM + d] * fcW[d * 3 + o];
  out[idx] = s + fcb[o];
}

// =========================================================================
static inline int blocksForWaves(int waves) { return (waves + 7) / 8; }

extern "C" void kernel_launch(void* const* d_in, const int* in_sizes, int n_in,
                              void* d_out, int out_size, void* d_ws,
                              size_t ws_size, hipStream_t stream) {
  const float* x = (const float*)d_in[0];
  const float* embW = (const float*)d_in[1];
  const float* embB = (const float*)d_in[2];
  const float* pcW = (const float*)d_in[3];
  const float* pcB = (const float*)d_in[4];
  const float* lnG = (const float*)d_in[5];
  const float* lnB = (const float*)d_in[6];
  const float* ipW = (const float*)d_in[7];
  const float* cW = (const float*)d_in[8];
  const float* cB = (const float*)d_in[9];
  const float* xpW = (const float*)d_in[10];
  const float* dtW = (const float*)d_in[11];
  const float* dtB = (const float*)d_in[12];
  const float* Alog = (const float*)d_in[13];
  const float* Dsk = (const float*)d_in[14];
  const float* opW = (const float*)d_in[15];
  const float* fcW = (const float*)d_in[16];
  const float* fcB = (const float*)d_in[17];
  float* out = (float*)d_out;

  float* ws = (float*)d_ws;
  float* h = ws;                              // MLEN*DM
  float* hn = h + (size_t)MLEN * DM;          // MLEN*DM
  float* xz = hn + (size_t)MLEN * DM;         // MLEN*1024
  float* u = xz + (size_t)MLEN * (2 * DI);    // MLEN*DI
  float* dbc = u + (size_t)MLEN * DI;         // MLEN*48
  float* dt = dbc + (size_t)MLEN * 48;        // MLEN*DI (doubles as gated y)
  float* pooled = dt + (size_t)MLEN * DI;     // B_*DM

  // embed -> hn ; posconv(hn) -> h
  embed_kernel<<<MLEN, 256, 0, stream>>>(x, embW, embB, hn);
  {
    int waves = (MLEN / 64) * (DM / 32);  // BM=4, BN=2
    posconv_kernel<4, 2><<<blocksForWaves(waves), 256, 0, stream>>>(hn, pcW,
                                                                    pcB, h);
  }

  for (int i = 0; i < 3; ++i) {
    ln_kernel<<<(MLEN + 7) / 8, 256, 0, stream>>>(h, lnG + i * DM, lnB + i * DM,
                                                  hn);
    {  // in_proj: (MLEN,256) @ (1024,256)^T -> xz
      int waves = (MLEN / 64) * ((2 * DI) / 32);
      gemm_wmma<4, 2, 0, false, false>
          <<<blocksForWaves(waves), 256, 0, stream>>>(
              hn, DM, ipW + (size_t)i * 2 * DI * DM, DM, nullptr, xz, 2 * DI,
              MLEN, 2 * DI, DM);
    }
    conv_silu_kernel<<<(MLEN * DI + 255) / 256, 256, 0, stream>>>(
        xz, cW + i * DI * 2, cB + i * DI, u);
    {  // x_proj: (MLEN,512) @ (48,512)^T -> dbc   (BN=3 covers all 48 cols)
      int waves = (MLEN / 64) * (48 / 48);
      gemm_wmma<4, 3, 0, false, false>
          <<<blocksForWaves(waves), 256, 0, stream>>>(
              u, DI, xpW + (size_t)i * 48 * DI, DI, nullptr, dbc, 48, MLEN, 48,
              DI);
    }
    {  // dt_proj + bias + softplus: (MLEN,16) @ (512,16)^T -> dt
      int waves = (MLEN / 64) * (DI / 32);
      gemm_wmma<4, 2, 1, false, true><<<blocksForWaves(waves), 256, 0, stream>>>(
          dbc, 48, dtW + (size_t)i * DI * DR, DR, dtB + i * DI, dt, DI, MLEN,
          DI, DR);
    }
    scan_kernel<<<(B_ * DI) / 256, 256, 0, stream>>>(
        u, dt, dbc, xz, Alog + (size_t)i * DI * DS_, Dsk + i * DI);
    {  // out_proj with residual: h += (MLEN,512) @ (256,512)^T
      int waves = (MLEN / 64) * (DM / 32);
      gemm_wmma<4, 2, 0, true, false><<<blocksForWaves(waves), 256, 0, stream>>>(
          dt, DI, opW + (size_t)i * DM * DI, DI, nullptr, h, DM, MLEN, DM, DI);
    }
  }

  pool_kernel<<<(B_ * DM + 255) / 256, 256, 0, stream>>>(h, pooled);
  fc_kernel<<<1, 32, 0, stream>>>(pooled, fcW, fcB, out);
}